// CARAFE_9337258901626
// MI455X (gfx1250) — compile-verified
//
#include <hip/hip_runtime.h>

// CARAFE pipeline for gfx1250 (MI455X), fp32 end-to-end using V_WMMA_F32_16X16X4_F32,
// with Tensor Data Mover (tensor_load_to_lds) staging of the B tiles for the 1x1 convs.
// N=4, INC=256, CM=64, ENC_OUT=100 (25 taps x 4 subpixels), H=W=64, upx2 -> 128x128.

typedef __attribute__((ext_vector_type(2))) float        v2f;
typedef __attribute__((ext_vector_type(8))) float        v8f;
typedef __attribute__((ext_vector_type(4))) unsigned int v4u;
typedef __attribute__((ext_vector_type(8))) int          v8i;
typedef __attribute__((ext_vector_type(4))) int          v4i;

#define HW    4096      // 64*64
#define HW2   16384     // 128*128

// ---------------------------------------------------------------------------
// TDM: async-load a (rows x 16) fp32 tile (row stride `stride` elements) from
// global into LDS.  D# built per CDNA5 ISA: group0 = {count=1, lds_addr,
// global_addr, type=2}; group1 = {data_size=4B, tensor_dim0/1, tile_dim0=16,
// tile_dim1=rows, tensor_dim0_stride}.  Remaining groups zero (2D tensor,
// no iteration / multicast / gather).  This toolchain exposes the 6-arg
// builtin form: (v4u, v8i, v4i, v4i, v8i, cpol).
// Caller must be wave-uniform; completion via s_wait_tensorcnt.
// ---------------------------------------------------------------------------
__device__ __forceinline__ void tdm_load_cols16(float* lds, const float* gsrc,
                                                unsigned rows, unsigned stride) {
  unsigned long long ga = (unsigned long long)(uintptr_t)gsrc;
  v4u g0;
  g0[0] = 1u;                                            // count=1 (user mode)
  g0[1] = (unsigned)(uintptr_t)lds;                      // LDS byte offset (flat addr low 32)
  g0[2] = (unsigned)ga;                                  // global_addr[31:0]
  g0[3] = (unsigned)((ga >> 32) & 0x01ffffffull) | 0x80000000u; // addr[56:32] | type=2<<30
  v8i g1;
  g1[0] = 0x00020000;                                    // data_size = 2 -> 4 bytes
  g1[1] = (int)((stride & 0xffffu) << 16);               // tensor_dim0[15:0]
  g1[2] = (int)(((stride >> 16) & 0xffffu)               // tensor_dim0[31:16]
                | ((rows & 0xffffu) << 16));             // tensor_dim1[15:0]
  g1[3] = (int)(16u << 16);                              // tensor_dim1[31:16]=0 | tile_dim0=16
  g1[4] = (int)(rows & 0xffffu);                         // tile_dim1=rows, tile_dim2=0
  g1[5] = (int)stride;                                   // tensor_dim0_stride[31:0]
  g1[6] = 0;                                             // stride hi | dim1_stride lo
  g1[7] = 0;
  v4i gz4 = {0, 0, 0, 0};
  v8i gz8 = {0, 0, 0, 0, 0, 0, 0, 0};
  __builtin_amdgcn_tensor_load_to_lds(g0, g1, gz4, gz4, gz8, 0);
}

// ---------------------------------------------------------------------------
// Kernel 1: 1x1 "down" conv.  out[n,co,hw] = sum_ci W[co,ci]*x[n,ci,hw] + b[co]
// GEMM M=64 (4 tiles), K=256 (64 wmma), N=16384 pixels.
// Block = 128 threads = 4 waves; wave w -> M-tile w; block -> one 16-pixel tile.
// B tile (256x16 fp32, 16KB) staged to LDS by one TDM op per block.
// ---------------------------------------------------------------------------
__global__ __launch_bounds__(128) void k_down(const float* __restrict__ x,
                                              const float* __restrict__ w,
                                              const float* __restrict__ bias,
                                              float* __restrict__ out) {
  __shared__ __align__(16) float bs[256 * 16];
  const int lane = threadIdx.x & 31;
  const int wv   = threadIdx.x >> 5;     // M-tile 0..3
  const int l15  = lane & 15;
  const int hi   = lane >> 4;            // lane half
  const int koff = hi * 2;               // fp32 A/B K sub-offset per ISA layout
  const int n    = blockIdx.x >> 8;
  const int p0   = (blockIdx.x & 255) << 4;

  if (wv == 0) {                                  // wave-uniform: TDM ignores EXEC
    tdm_load_cols16(bs, x + (size_t)n * 256 * HW + p0, 256u, (unsigned)HW);
    __builtin_amdgcn_s_wait_tensorcnt(0);
  }
  __syncthreads();

  const int row   = wv * 16 + l15;                         // output channel
  const float* ap = w + (size_t)row * 256 + koff;          // weight row

  v8f acc = {0.f,0.f,0.f,0.f,0.f,0.f,0.f,0.f};
#pragma unroll 8
  for (int k = 0; k < 256; k += 4) {
    v2f a = *(const v2f*)(ap + k);
    v2f b;                                                 // conflict-free: halves hit
    b.x = bs[(k + koff    ) * 16 + l15];                   // disjoint 16-bank groups
    b.y = bs[(k + koff + 1) * 16 + l15];
    acc = __builtin_amdgcn_wmma_f32_16x16x4_f32(false, a, false, b,
                                                (short)0, acc, false, false);
  }
  float* on = out + (size_t)n * 64 * HW;
#pragma unroll
  for (int i = 0; i < 8; ++i) {
    int r = wv * 16 + i + hi * 8;
    on[(size_t)r * HW + p0 + l15] = acc[i] + bias[r];
  }
}

// ---------------------------------------------------------------------------
// Kernel 2: 3x3 "encoder" conv, pad 1.  im2col GEMM M=100(->7 tiles), K=576.
// K is permuted tap-major: k' = t*64 + ci, so the (kh,kw,bounds) decode is
// uniform per 64-chunk and hoisted; inner loop = 2 strided loads + 7 WMMAs.
// Block = 128 threads = 4 waves; each wave owns one 16-pixel tile.
// ---------------------------------------------------------------------------
__global__ __launch_bounds__(128) void k_enc(const float* __restrict__ feat,
                                             const float* __restrict__ w,
                                             const float* __restrict__ bias,
                                             float* __restrict__ out) {
  const int lane = threadIdx.x & 31;
  const int wv   = threadIdx.x >> 5;
  const int l15  = lane & 15;
  const int hi   = lane >> 4;
  const int koff = hi * 2;
  const int tile = blockIdx.x * 4 + wv;     // 0..1023
  const int n    = tile >> 8;
  const int p0   = (tile & 255) << 4;       // 16 | 64 -> tile stays on one row
  const int h    = p0 >> 6;
  const int wc   = (p0 & 63) + l15;         // pixel x for B lanes

  const float* dn = feat + (size_t)n * 64 * HW;

  // Per-M-tile weight row base + zero-mask for the padded rows 100..111.
  float        msk[7];
  const float* wrow[7];
#pragma unroll
  for (int m = 0; m < 7; ++m) {
    int row = m * 16 + l15;
    msk[m]  = (row < 100) ? 1.f : 0.f;
    wrow[m] = w + (size_t)((row < 100) ? row : 0) * 576;
  }

  v8f acc[7];
#pragma unroll
  for (int m = 0; m < 7; ++m) acc[m] = (v8f){0.f,0.f,0.f,0.f,0.f,0.f,0.f,0.f};

#pragma unroll
  for (int t = 0; t < 9; ++t) {             // tap index: kh*3+kw
    const int  kh = t / 3, kw = t - kh * 3;
    const int  hh = h  + kh - 1;
    const int  ww = wc + kw - 1;
    const bool ok = ((unsigned)hh < 64u) & ((unsigned)ww < 64u);
    const float* drow = dn + hh * 64 + ww;  // + ci*HW (deref guarded by ok)
    for (int ci = 0; ci < 64; ci += 4) {
      const int c0 = ci + koff;
      v2f b;
      b.x = ok ? drow[(size_t)(c0    ) * HW] : 0.f;
      b.y = ok ? drow[(size_t)(c0 + 1) * HW] : 0.f;
#pragma unroll
      for (int m = 0; m < 7; ++m) {
        const float* ap = wrow[m] + (size_t)c0 * 9 + t;   // stride-9 channel walk
        v2f a; a.x = ap[0] * msk[m]; a.y = ap[9] * msk[m];
        acc[m] = __builtin_amdgcn_wmma_f32_16x16x4_f32(false, a, false, b,
                                                       (short)0, acc[m], false, false);
      }
    }
  }
  float* on = out + (size_t)n * 100 * HW;
#pragma unroll
  for (int m = 0; m < 7; ++m)
#pragma unroll
    for (int i = 0; i < 8; ++i) {
      int r = m * 16 + i + hi * 8;
      if (r < 100) on[(size_t)r * HW + p0 + l15] = acc[m][i] + bias[r];
    }
}

// ---------------------------------------------------------------------------
// Kernel 3: softmax over the 25 taps for each (n, subpixel d, hw).
// Channel layout: c = k*4 + d  (torch pixel_shuffle order), stride 4*HW.
// ---------------------------------------------------------------------------
__global__ __launch_bounds__(256) void k_softmax(const float* __restrict__ enc,
                                                 float* __restrict__ kern) {
  int tid = blockIdx.x * 256 + threadIdx.x;    // 65536 = 4*4*4096
  int n   = tid >> 14;
  int rem = tid & 16383;
  int d   = rem >> 12;
  int hw  = rem & 4095;
  const float* p = enc + (size_t)n * 100 * HW + (size_t)d * HW + hw;
  float v[25];
  float mx = -3.4e38f;
#pragma unroll
  for (int k = 0; k < 25; ++k) { v[k] = p[(size_t)k * 4 * HW]; mx = v[k] > mx ? v[k] : mx; }
  float s = 0.f;
#pragma unroll
  for (int k = 0; k < 25; ++k) { v[k] = expf(v[k] - mx); s += v[k]; }
  float inv = 1.f / s;
  float* q = kern + (size_t)n * 100 * HW + (size_t)d * HW + hw;
#pragma unroll
  for (int k = 0; k < 25; ++k) q[(size_t)k * 4 * HW] = v[k] * inv;
}

// ---------------------------------------------------------------------------
// Kernel 4: feature reassembly + fused pixel-shuffle.
// up[n,c,2h+i2,2w+j2] = sum_{i,j} x[n,c,h+i-2,w+j-2] * kern[n,(i*5+j)*4+d,h,w]
// Block = one (n,h) row: threads = (w:64) x (c-phase:4); per-row kernel weights
// staged in LDS (layout [k4][w] -> conflict-free lane-consecutive reads).
// ---------------------------------------------------------------------------
__global__ __launch_bounds__(256) void k_reassemble(const float* __restrict__ x,
                                                    const float* __restrict__ kern,
                                                    float* __restrict__ up) {
  __shared__ float ks[100 * 64];               // 25.6 KB of the 320 KB WGP LDS
  const int n  = blockIdx.x >> 6;
  const int h  = blockIdx.x & 63;
  const int wq = threadIdx.x & 63;
  const int c4 = threadIdx.x >> 6;

  const float* kb = kern + (size_t)n * 100 * HW + h * 64;
  for (int idx = threadIdx.x; idx < 6400; idx += 256)
    ks[idx] = kb[(size_t)(idx >> 6) * HW + (idx & 63)];
  __syncthreads();

  const float* xn = x  + (size_t)n * 256 * HW;
  float*       un = up + (size_t)n * 256 * HW2;

  for (int c = c4; c < 256; c += 4) {
    const float* xc = xn + (size_t)c * HW;
    float a0 = 0.f, a1 = 0.f, a2 = 0.f, a3 = 0.f;
#pragma unroll
    for (int i = 0; i < 5; ++i) {
      int hh = h + i - 2;
      if ((unsigned)hh < 64u) {
#pragma unroll
        for (int j = 0; j < 5; ++j) {
          int ww = wq + j - 2;
          float px = ((unsigned)ww < 64u) ? xc[hh * 64 + ww] : 0.f;
          int k = i * 5 + j;
          a0 += px * ks[(k * 4 + 0) * 64 + wq];
          a1 += px * ks[(k * 4 + 1) * 64 + wq];
          a2 += px * ks[(k * 4 + 2) * 64 + wq];
          a3 += px * ks[(k * 4 + 3) * 64 + wq];
        }
      }
    }
    float* uc = un + (size_t)c * HW2;
    *(float2*)(uc + (size_t)(2 * h    ) * 128 + 2 * wq) = make_float2(a0, a1);
    *(float2*)(uc + (size_t)(2 * h + 1) * 128 + 2 * wq) = make_float2(a2, a3);
  }
}

// ---------------------------------------------------------------------------
// Kernel 5: final 1x1 conv on upsampled features.
// GEMM M=256 (16 tiles), K=256 (64 wmma), N=65536 pixels.
// Block = 256 threads = 8 waves; wave wv owns M-tiles {wv, wv+8}, B tile
// (256x16 fp32, 16KB) staged to LDS once per block via TDM.
// ---------------------------------------------------------------------------
__global__ __launch_bounds__(256) void k_out(const float* __restrict__ up,
                                             const float* __restrict__ w,
                                             const float* __restrict__ bias,
                                             float* __restrict__ out) {
  __shared__ __align__(16) float bs[256 * 16];
  const int lane = threadIdx.x & 31;
  const int wv   = threadIdx.x >> 5;        // 0..7
  const int l15  = lane & 15;
  const int hi   = lane >> 4;
  const int koff = hi * 2;
  const int n    = blockIdx.x >> 10;
  const int p0   = (blockIdx.x & 1023) << 4;

  if (wv == 0) {                                  // wave-uniform: TDM ignores EXEC
    tdm_load_cols16(bs, up + (size_t)n * 256 * HW2 + p0, 256u, (unsigned)HW2);
    __builtin_amdgcn_s_wait_tensorcnt(0);
  }
  __syncthreads();

  const float* a0p = w + (size_t)(wv * 16 + l15) * 256 + koff;
  const float* a1p = a0p + (size_t)128 * 256;      // M-tile +8 -> +128 rows

  v8f acc0 = {0.f,0.f,0.f,0.f,0.f,0.f,0.f,0.f};
  v8f acc1 = {0.f,0.f,0.f,0.f,0.f,0.f,0.f,0.f};
#pragma unroll 4
  for (int k = 0; k < 256; k += 4) {
    v2f b;
    b.x = bs[(k + koff    ) * 16 + l15];
    b.y = bs[(k + koff + 1) * 16 + l15];
    v2f a0 = *(const v2f*)(a0p + k);
    v2f a1 = *(const v2f*)(a1p + k);
    acc0 = __builtin_amdgcn_wmma_f32_16x16x4_f32(false, a0, false, b,
                                                 (short)0, acc0, false, false);
    acc1 = __builtin_amdgcn_wmma_f32_16x16x4_f32(false, a1, false, b,
                                                 (short)0, acc1, false, false);
  }
  float* on = out + (size_t)n * 256 * HW2;
#pragma unroll
  for (int i = 0; i < 8; ++i) {
    int r0 = wv * 16 + i + hi * 8;
    on[(size_t)r0 * HW2 + p0 + l15] = acc0[i] + bias[r0];
    int r1 = r0 + 128;
    on[(size_t)r1 * HW2 + p0 + l15] = acc1[i] + bias[r1];
  }
}

// ---------------------------------------------------------------------------
extern "C" void kernel_launch(void* const* d_in, const int* in_sizes, int n_in,
                              void* d_out, int out_size, void* d_ws, size_t ws_size,
                              hipStream_t stream) {
  const float* x      = (const float*)d_in[0];   // (4,256,64,64)
  const float* w_down = (const float*)d_in[1];   // (64,256,1,1)
  const float* b_down = (const float*)d_in[2];   // (64)
  const float* w_enc  = (const float*)d_in[3];   // (100,64,3,3)
  const float* b_enc  = (const float*)d_in[4];   // (100)
  const float* w_out  = (const float*)d_in[5];   // (256,256,1,1)
  const float* b_out  = (const float*)d_in[6];   // (256)
  float* out = (float*)d_out;                    // (4,256,128,128)

  float* ws     = (float*)d_ws;
  float* t_down = ws;                            //  4*64*4096   = 1,048,576 f
  float* t_enc  = t_down + (size_t)4 * 64  * HW; //  4*100*4096  = 1,638,400 f
  float* t_kern = t_enc  + (size_t)4 * 100 * HW; //  same size
  float* t_up   = t_kern + (size_t)4 * 100 * HW; //  4*256*16384 = 16,777,216 f

  k_down      <<<1024, 128, 0, stream>>>(x, w_down, b_down, t_down);
  k_enc       <<< 256, 128, 0, stream>>>(t_down, w_enc, b_enc, t_enc);
  k_softmax   <<< 256, 256, 0, stream>>>(t_enc, t_kern);
  k_reassemble<<< 256, 256, 0, stream>>>(x, t_kern, t_up);
  k_out       <<<4096, 256, 0, stream>>>(t_up, w_out, b_out, out);
}